// Attention_89163521065175
// MI455X (gfx1250) — compile-verified
//
#include <hip/hip_runtime.h>
#include <hip/hip_bf16.h>
#include <math.h>

typedef float v2f __attribute__((ext_vector_type(2)));
typedef float v8f __attribute__((ext_vector_type(8)));

#define D_FEAT 32

// ---------------------------------------------------------------------------
// Kernel 0: zero the per-node softmax denominators and the output buffer.
// ---------------------------------------------------------------------------
__global__ void attn_init_kernel(float* __restrict__ sums, float* __restrict__ out,
                                 int n_nodes, int out_elems) {
    int i = blockIdx.x * blockDim.x + threadIdx.x;
    int stride = gridDim.x * blockDim.x;
    for (int j = i; j < out_elems; j += stride) out[j] = 0.0f;
    for (int j = i; j < n_nodes; j += stride) sums[j] = 0.0f;
}

// ---------------------------------------------------------------------------
// Kernel 1: per-edge score via f32 WMMA.
// One wave32 handles a tile of 16 edges. The 64-long dot product
// [x_j+e_ij, x_i] . W is computed as 16 chained V_WMMA_F32_16X16X4_F32 steps
// (K=4 each). B carries W only in column N=0, so the 16 edge scores end up in
// column 0 of the 16x16 f32 accumulator: lane 0 holds M=0..7 in c[0..7],
// lane 16 holds M=8..15 (per the documented C/D VGPR layout).
//
// W is staged in LDS once per block; B fragments are built with unconditional
// LDS broadcast reads + register selects (v_cndmask), avoiding the predicated
// global loads (saveexec/xcnt churn) the ternary-on-load version produced.
// ---------------------------------------------------------------------------
__global__ void attn_score_kernel(const float* __restrict__ x_i,
                                  const float* __restrict__ x_j,
                                  const float* __restrict__ e_ij,
                                  const float* __restrict__ W,
                                  const float* __restrict__ bias,
                                  const int* __restrict__ index,
                                  float* __restrict__ p_out,
                                  float* __restrict__ sums,
                                  int E) {
    __shared__ float Wsh[64];
    if (threadIdx.x < 64) Wsh[threadIdx.x] = W[threadIdx.x];
    __syncthreads();                        // all threads reach this exactly once

    const int lane = threadIdx.x & 31;
    const int half = lane >> 4;   // 0: K pair {0,1}, 1: K pair {2,3} (A/B layout)
    const int m    = lane & 15;   // edge-in-tile (A rows) / N column (B,C,D)

    const int wave     = (blockIdx.x * blockDim.x + threadIdx.x) >> 5;
    const int numTiles = (E + 15) >> 4;
    if (wave >= numTiles) return;           // wave-uniform exit: EXEC stays all-1s

    const int base = wave << 4;
    int e = base + m;
    if (e >= E) e = E - 1;                  // clamp loads for a partial tail tile

    const float* xj_row  = x_j  + (size_t)e * D_FEAT;
    const float* eij_row = e_ij + (size_t)e * D_FEAT;
    const float* xi_row  = x_i  + (size_t)e * D_FEAT;

    const bool isN0 = (m == 0);             // lanes 0 and 16 supply B column 0

    v8f c = {};                              // f32 16x16 accumulator (8 VGPRs)

#pragma unroll
    for (int t = 0; t < 16; ++t) {
        const int k = 4 * t + 2 * half;      // feature index in [0,64)
        // A fragment: lane holds A[m][2*half], A[m][2*half+1] of this K=4 slab.
        v2f a;
        if (t < 8) {                         // features 0..31 -> q = x_j + e_ij
            a.x = xj_row[k]     + eij_row[k];
            a.y = xj_row[k + 1] + eij_row[k + 1];
        } else {                             // features 32..63 -> x_i
            a.x = xi_row[k - 32];
            a.y = xi_row[k - 32 + 1];
        }
        // B fragment: B[kk][n] = W[4t+kk] if n==0 else 0.
        // Unconditional LDS broadcast reads, then register select.
        const float w0 = Wsh[k];
        const float w1 = Wsh[k + 1];
        v2f bb;
        bb.x = isN0 ? w0 : 0.0f;
        bb.y = isN0 ? w1 : 0.0f;

        c = __builtin_amdgcn_wmma_f32_16x16x4_f32(
                /*neg_a=*/false, a, /*neg_b=*/false, bb,
                /*c_mod=*/(short)0, c, /*reuse_a=*/false, /*reuse_b=*/false);
    }

    // Column N=0 lives in lanes 0 (M=0..7) and 16 (M=8..15).
    if (isN0) {
        const float bv = bias[0];
        const int mbase = half * 8;
#pragma unroll
        for (int r = 0; r < 8; ++r) {
            int ee = base + mbase + r;
            if (ee < E) {
                float s  = tanhf(c[r] + bv);
                float pe = expf(s);          // shift-free softmax: tanh bounded
                p_out[ee] = pe;
                atomicAdd(&sums[index[ee]], pe);
            }
        }
    }
}

// ---------------------------------------------------------------------------
// Kernel 2: normalized weighted scatter-sum of msg.
// index is sorted -> accumulate runs of equal node id in registers, flush with
// one atomicAdd per run boundary. Lane = feature (D_FEAT == wave32 width).
// ---------------------------------------------------------------------------
__global__ void attn_agg_kernel(const float* __restrict__ msg,
                                const float* __restrict__ p,
                                const float* __restrict__ sums,
                                const int* __restrict__ index,
                                float* __restrict__ out,
                                int E) {
    const int lane  = threadIdx.x & 31;
    const int wave  = (blockIdx.x * blockDim.x + threadIdx.x) >> 5;
    const int nWave = (gridDim.x * blockDim.x) >> 5;

    const int CHUNK   = 256;
    const int nChunks = (E + CHUNK - 1) / CHUNK;

    for (int ch = wave; ch < nChunks; ch += nWave) {
        const int start = ch * CHUNK;
        const int end   = min(start + CHUNK, E);

        float acc = 0.0f;
        int cur = index[start];
        for (int e = start; e < end; ++e) {
            const int node = index[e];       // wave-uniform
            if (node != cur) {               // wave-uniform branch
                atomicAdd(&out[(size_t)cur * D_FEAT + lane], acc);
                acc = 0.0f;
                cur = node;
            }
            const float w = p[e] / (sums[node] + 1e-16f);
            acc += w * msg[(size_t)e * D_FEAT + lane];
        }
        atomicAdd(&out[(size_t)cur * D_FEAT + lane], acc);
    }
}

// ---------------------------------------------------------------------------
// Host launcher. Input order: msg, x_i, x_j, e_ij, W, b, index, num_nodes.
// Workspace: p[E] floats, sums[N] floats (~6.8 MB).
// ---------------------------------------------------------------------------
extern "C" void kernel_launch(void* const* d_in, const int* in_sizes, int n_in,
                              void* d_out, int out_size, void* d_ws, size_t ws_size,
                              hipStream_t stream) {
    const float* msg   = (const float*)d_in[0];
    const float* x_i   = (const float*)d_in[1];
    const float* x_j   = (const float*)d_in[2];
    const float* e_ij  = (const float*)d_in[3];
    const float* W     = (const float*)d_in[4];
    const float* bias  = (const float*)d_in[5];
    const int*   index = (const int*)d_in[6];

    const int E = in_sizes[6];          // number of edges
    const int N = out_size / D_FEAT;    // number of nodes

    float* out  = (float*)d_out;
    float* p    = (float*)d_ws;         // E floats
    float* sums = p + E;                // N floats

    // Pass 0: zero denominators + output.
    {
        int blocks = (out_size + 255) / 256;
        if (blocks > 65535) blocks = 65535;
        attn_init_kernel<<<blocks, 256, 0, stream>>>(sums, out, N, out_size);
    }

    // Pass 1: per-edge scores (wave per 16-edge tile, 8 waves per block).
    {
        int tiles  = (E + 15) / 16;
        int blocks = (tiles + 7) / 8;
        attn_score_kernel<<<blocks, 256, 0, stream>>>(x_i, x_j, e_ij, W, bias,
                                                      index, p, sums, E);
    }

    // Pass 2: weighted scatter-sum (wave per 256-edge chunk).
    {
        int chunks = (E + 255) / 256;
        int blocks = (chunks + 7) / 8;
        if (blocks > 4096) blocks = 4096;
        attn_agg_kernel<<<blocks, 256, 0, stream>>>(msg, p, sums, index, out, E);
    }
}